// RecommenderModel_69518340653459
// MI455X (gfx1250) — compile-verified
//
#include <hip/hip_runtime.h>
#include <hip/hip_bf16.h>

typedef __attribute__((ext_vector_type(2))) float v2f;
typedef __attribute__((ext_vector_type(8))) float v8f;

#define FDIM 128

// ---------------------------------------------------------------------------
// Fused GEMM:  Out[N,128] = op( A1[N,128] @ W1[128,128] + (A2*rdeg)[N,128] @ W2 )
// op = ReLU if relu!=0.  Mean-aggregation division is folded in via per-row
// scale rdeg = 1/max(deg[row],1) applied to the A2 fragment.
// One block = 16 output rows; 8 waves, each wave owns one 16-col tile.
// Uses V_WMMA_F32_16X16X4_F32 (full f32 precision, K=4 per instruction).
// Out may alias A2 (each block only touches its own 16 rows; __syncthreads()
// separates all A2 reads from the stores).
// ---------------------------------------------------------------------------
__global__ __launch_bounds__(256)
void fused_gemm_wmma(const float* __restrict__ A1,
                     const float* A2,
                     const float* __restrict__ deg,
                     const float* __restrict__ W1,
                     const float* __restrict__ W2,
                     float* Out,
                     int relu)
{
    const int lane  = threadIdx.x & 31;
    const int wave  = threadIdx.x >> 5;        // 0..7 -> column tile
    const int mbase = blockIdx.x << 4;         // 16 rows per block
    const int nbase = wave << 4;               // 16 cols per wave

    const int mrow  = lane & 15;
    const int khalf = lane >> 4;               // 0 or 1 (K-pair select, ISA A layout)
    const int gm    = mbase + mrow;
    const int ncol  = nbase + (lane & 15);

    const float d    = deg[gm];
    const float rdeg = 1.0f / fmaxf(d, 1.0f);

    const float* a1p = A1 + (size_t)gm * FDIM;
    const float* a2p = A2 + (size_t)gm * FDIM;

    v8f acc = {0.f, 0.f, 0.f, 0.f, 0.f, 0.f, 0.f, 0.f};

#pragma unroll 4
    for (int k0 = 0; k0 < FDIM; k0 += 4) {
        const int kb = k0 + khalf * 2;
        // A fragments: lane holds (row=gm, K=kb..kb+1) -> contiguous b64 load
        v2f a1 = *(const v2f*)(a1p + kb);
        v2f a2 = *(const v2f*)(a2p + kb);
        a2.x *= rdeg;
        a2.y *= rdeg;
        // B fragments: lane holds (K=kb..kb+1, col=ncol)
        v2f b1, b2;
        b1.x = W1[(size_t)(kb    ) * FDIM + ncol];
        b1.y = W1[(size_t)(kb + 1) * FDIM + ncol];
        b2.x = W2[(size_t)(kb    ) * FDIM + ncol];
        b2.y = W2[(size_t)(kb + 1) * FDIM + ncol];
        acc = __builtin_amdgcn_wmma_f32_16x16x4_f32(false, a1, false, b1,
                                                    (short)0, acc, false, false);
        acc = __builtin_amdgcn_wmma_f32_16x16x4_f32(false, a2, false, b2,
                                                    (short)0, acc, false, false);
    }

    if (relu) {
#pragma unroll
        for (int r = 0; r < 8; ++r) acc[r] = fmaxf(acc[r], 0.0f);
    }

    __syncthreads();   // make Out-aliases-A2 safe: all reads done block-wide

    // C/D layout: VGPR r -> M = khalf*8 + r, N = ncol
#pragma unroll
    for (int r = 0; r < 8; ++r) {
        const int row = mbase + khalf * 8 + r;
        Out[(size_t)row * FDIM + ncol] = acc[r];
    }
}

// ---------------------------------------------------------------------------
// Degree counting: one thread per u->i edge; its mirror i->u edge gives the
// user degree, so both histograms are built in one pass.
// ---------------------------------------------------------------------------
__global__ __launch_bounds__(256)
void count_degrees(const int* __restrict__ usrc,
                   const int* __restrict__ idst,
                   float* __restrict__ deg_u,
                   float* __restrict__ deg_i,
                   int nedges)
{
    const int e = blockIdx.x * blockDim.x + threadIdx.x;
    if (e >= nedges) return;
    atomicAdd(deg_u + usrc[e], 1.0f);
    atomicAdd(deg_i + idst[e], 1.0f);
}

// ---------------------------------------------------------------------------
// Edge scatter-add: one wave per edge; 32 lanes x float4 = 128 floats.
// Coalesced 512B gather of the source row, 4 f32 global atomics per lane
// into the destination row.
// ---------------------------------------------------------------------------
__global__ __launch_bounds__(256)
void scatter_add_rows(const float* __restrict__ X,
                      const int* __restrict__ src,
                      const int* __restrict__ dst,
                      float* __restrict__ agg,
                      int nedges)
{
    const int lane = threadIdx.x & 31;
    const int e    = blockIdx.x * 8 + (threadIdx.x >> 5);
    if (e >= nedges) return;
    const int s = src[e];
    const int t = dst[e];
    const float4 v = *(const float4*)(X + (size_t)s * FDIM + lane * 4);
    float* p = agg + (size_t)t * FDIM + lane * 4;
    atomicAdd(p + 0, v.x);
    atomicAdd(p + 1, v.y);
    atomicAdd(p + 2, v.z);
    atomicAdd(p + 3, v.w);
}

// ---------------------------------------------------------------------------
// Pair scoring: one wave per label edge; float4 loads of both embeddings,
// wave32 __shfl_xor butterfly reduction of the dot product.
// ---------------------------------------------------------------------------
__global__ __launch_bounds__(256)
void score_pairs(const float* __restrict__ zu,
                 const float* __restrict__ zi,
                 const int* __restrict__ rowi,
                 const int* __restrict__ coli,
                 float* __restrict__ out,
                 int npairs)
{
    const int lane = threadIdx.x & 31;
    const int e    = blockIdx.x * 8 + (threadIdx.x >> 5);
    if (e >= npairs) return;
    const float4 a = *(const float4*)(zu + (size_t)rowi[e] * FDIM + lane * 4);
    const float4 b = *(const float4*)(zi + (size_t)coli[e] * FDIM + lane * 4);
    float p = a.x * b.x + a.y * b.y + a.z * b.z + a.w * b.w;
#pragma unroll
    for (int off = 16; off > 0; off >>= 1) p += __shfl_xor(p, off, 32);
    if (lane == 0) out[e] = p;
}

// ---------------------------------------------------------------------------
extern "C" void kernel_launch(void* const* d_in, const int* in_sizes, int n_in,
                              void* d_out, int out_size, void* d_ws, size_t ws_size,
                              hipStream_t stream)
{
    const float* x_user    = (const float*)d_in[0];
    const float* x_item    = (const float*)d_in[1];
    const int*   ei_ui_src = (const int*)d_in[2];
    const int*   ei_ui_dst = (const int*)d_in[3];
    const int*   ei_iu_src = (const int*)d_in[4];
    const int*   ei_iu_dst = (const int*)d_in[5];
    const int*   rowi      = (const int*)d_in[6];
    const int*   coli      = (const int*)d_in[7];
    const float* Wsu1 = (const float*)d_in[8];
    const float* Wsi1 = (const float*)d_in[9];
    const float* Wui1 = (const float*)d_in[10];
    const float* Wiu1 = (const float*)d_in[11];
    const float* Wsu2 = (const float*)d_in[12];
    const float* Wsi2 = (const float*)d_in[13];
    const float* Wui2 = (const float*)d_in[14];
    const float* Wiu2 = (const float*)d_in[15];

    const int NU = in_sizes[0] / FDIM;   // 200000
    const int NI = in_sizes[1] / FDIM;   // 100000
    const int E  = in_sizes[2];          // 2000000
    const int L  = in_sizes[6];          // 500000

    // Workspace layout (floats): agg_u | agg_i | h_u | h_i | deg_u | deg_i
    float* ws = (float*)d_ws;
    size_t o = 0;
    float* agg_u = ws + o; o += (size_t)NU * FDIM;
    float* agg_i = ws + o; o += (size_t)NI * FDIM;
    float* h_u   = ws + o; o += (size_t)NU * FDIM;
    float* h_i   = ws + o; o += (size_t)NI * FDIM;
    float* deg_u = ws + o; o += (size_t)NU;
    float* deg_i = ws + o; o += (size_t)NI;

    const size_t bytes_u = (size_t)NU * FDIM * sizeof(float);
    const size_t bytes_i = (size_t)NI * FDIM * sizeof(float);

    hipMemsetAsync(agg_u, 0, bytes_u, stream);
    hipMemsetAsync(agg_i, 0, bytes_i, stream);
    hipMemsetAsync(deg_u, 0, (size_t)NU * sizeof(float), stream);
    hipMemsetAsync(deg_i, 0, (size_t)NI * sizeof(float), stream);

    // degrees (identical for both layers)
    count_degrees<<<(E + 255) / 256, 256, 0, stream>>>(ei_ui_src, ei_ui_dst,
                                                       deg_u, deg_i, E);

    // ---- layer 1 aggregation ----
    scatter_add_rows<<<(E + 7) / 8, 256, 0, stream>>>(x_user, ei_ui_src, ei_ui_dst, agg_i, E);
    scatter_add_rows<<<(E + 7) / 8, 256, 0, stream>>>(x_item, ei_iu_src, ei_iu_dst, agg_u, E);

    // ---- layer 1 fused GEMMs + ReLU ----
    fused_gemm_wmma<<<NU / 16, 256, 0, stream>>>(x_user, agg_u, deg_u, Wsu1, Wiu1, h_u, 1);
    fused_gemm_wmma<<<NI / 16, 256, 0, stream>>>(x_item, agg_i, deg_i, Wsi1, Wui1, h_i, 1);

    // ---- layer 2 aggregation ----
    hipMemsetAsync(agg_u, 0, bytes_u, stream);
    hipMemsetAsync(agg_i, 0, bytes_i, stream);
    scatter_add_rows<<<(E + 7) / 8, 256, 0, stream>>>(h_u, ei_ui_src, ei_ui_dst, agg_i, E);
    scatter_add_rows<<<(E + 7) / 8, 256, 0, stream>>>(h_i, ei_iu_src, ei_iu_dst, agg_u, E);

    // ---- layer 2 fused GEMMs (no ReLU); z written in place over agg ----
    fused_gemm_wmma<<<NU / 16, 256, 0, stream>>>(h_u, agg_u, deg_u, Wsu2, Wiu2, agg_u, 0);
    fused_gemm_wmma<<<NI / 16, 256, 0, stream>>>(h_i, agg_i, deg_i, Wsi2, Wui2, agg_i, 0);

    // ---- scoring ----
    score_pairs<<<(L + 7) / 8, 256, 0, stream>>>(agg_u, agg_i, rowi, coli,
                                                 (float*)d_out, L);
}